// MultiGCN_39874476376591
// MI455X (gfx1250) — compile-verified
//
#include <hip/hip_runtime.h>
#include <stdint.h>

// ---------------- problem constants (match reference) ----------------
#define N_NODES 30000
#define N_EDGES 480000
#define N_GRAPH 32
#define FDIM    512
#define HDIM    128
#define EPSN    1e-5f

typedef float v2f __attribute__((ext_vector_type(2)));
typedef float v8f __attribute__((ext_vector_type(8)));

// ---------------- utility kernels ----------------
__global__ void k_fill(float* p, float v, int n) {
    int i = blockIdx.x * blockDim.x + threadIdx.x;
    if (i < n) p[i] = v;
}

// x0p[n, 0:512]=x, [512]=pLDDT, [513:516]=0 (pad so K%4==0)
__global__ void k_build_x0(const float* __restrict__ x, const float* __restrict__ plddt,
                           float* __restrict__ x0p) {
    int i = blockIdx.x * blockDim.x + threadIdx.x;
    if (i >= N_NODES * 516) return;
    int n = i / 516, c = i - n * 516;
    float v = 0.f;
    if (c < FDIM) v = x[n * FDIM + c];
    else if (c == FDIM) v = plddt[n];
    x0p[i] = v;
}

// batch is sorted: per-graph [start,end) node ranges; start[N_GRAPH]=N
__global__ void k_starts(const int* __restrict__ batch, int* __restrict__ start) {
    int n = blockIdx.x * blockDim.x + threadIdx.x;
    if (n >= N_NODES) return;
    if (n == 0) {
        for (int g = 0; g <= batch[0]; ++g) start[g] = 0;
    } else if (batch[n] != batch[n - 1]) {
        for (int g = batch[n - 1] + 1; g <= batch[n]; ++g) start[g] = n;
    }
    if (n == N_NODES - 1) {
        for (int g = batch[n] + 1; g <= N_GRAPH; ++g) start[g] = N_NODES;
    }
}

// deg[col] += 1 (deg pre-filled with 1.0 for the self loop)
__global__ void k_degree(const int* __restrict__ col, float* __restrict__ deg) {
    int e = blockIdx.x * blockDim.x + threadIdx.x;
    if (e < N_EDGES) atomicAdd(&deg[col[e]], 1.0f);
}

__global__ void k_rsqrt_inplace(float* p, int n) {
    int i = blockIdx.x * blockDim.x + threadIdx.x;
    if (i < n) p[i] = rsqrtf(p[i]);
}

// ---------------- GraphNorm ----------------
__global__ void k_graph_stats(const float* __restrict__ H, int ldh, int C,
                              const int* __restrict__ start, const float* __restrict__ gn_a,
                              float* __restrict__ amean, float* __restrict__ invstd) {
    int g = blockIdx.x;
    int c = blockIdx.y * blockDim.x + threadIdx.x;
    if (c >= C) return;
    int s = start[g], e = start[g + 1];
    float cnt = fmaxf((float)(e - s), 1.0f);
    float sum = 0.f;
    for (int n = s; n < e; ++n) sum += H[(size_t)n * ldh + c];
    float am = gn_a[c] * (sum / cnt);
    float s2 = 0.f;
    for (int n = s; n < e; ++n) { float d = H[(size_t)n * ldh + c] - am; s2 += d * d; }
    amean[g * C + c]  = am;
    invstd[g * C + c] = rsqrtf(s2 / cnt + EPSN);
}

// out[n, 0:Cpad] = g*(h-amean)*invstd + b  (zero in pad columns), row stride Cpad
__global__ void k_gn_apply(const float* __restrict__ H, int ldh, int C, int Cpad,
                           const int* __restrict__ batch,
                           const float* __restrict__ amean, const float* __restrict__ invstd,
                           const float* __restrict__ gw, const float* __restrict__ gb,
                           float* __restrict__ out) {
    long i = (long)blockIdx.x * blockDim.x + threadIdx.x;
    if (i >= (long)N_NODES * Cpad) return;
    int n = (int)(i / Cpad), c = (int)(i - (long)n * Cpad);
    float v = 0.f;
    if (c < C) {
        int g = batch[n];
        v = gw[c] * (H[(size_t)n * ldh + c] - amean[g * C + c]) * invstd[g * C + c] + gb[c];
    }
    out[i] = v;
}

// transpose+pad W [C,128] -> Wt [128, Cpad] (zero-padded K tail) so B fragments
// become single contiguous 8B loads per lane in the WMMA loop
__global__ void k_padWt(const float* __restrict__ W, int C, int Cpad, float* __restrict__ Wt) {
    int i = blockIdx.x * blockDim.x + threadIdx.x;      // i = c*Cpad + r (coalesced writes)
    if (i >= Cpad * HDIM) return;
    int c = i / Cpad, r = i - c * Cpad;
    Wt[i] = (r < C) ? W[r * HDIM + c] : 0.f;
}

// ---------------- fp32 WMMA GEMM: Y[N,128] = A[N,K4] @ Wt[128,K4]^T ----------------
// block = 256 threads = 8 waves; block owns a 16-row M tile; wave w owns N cols [16w,16w+16).
// A tile staged in LDS via CDNA5 async global->LDS DMA (ASYNCcnt path), stride ldss%64==4
// keeps the 16-row x {0,2}-K access pattern bank-conflict-free on 64 banks.
__global__ __launch_bounds__(256)
void k_gemm_f32_wmma(const float* __restrict__ A, const float* __restrict__ Wt,
                     float* __restrict__ Y, int K4, int ldss) {
    extern __shared__ float As[];
    const int m0 = blockIdx.x * 16;
    const int kchunks = K4 >> 2;                 // 16B chunks per row
    for (int i = threadIdx.x; i < 16 * kchunks; i += 256) {
        int r = i / kchunks;
        int c = (i - r * kchunks) << 2;
        uint32_t lds_addr = (uint32_t)(uintptr_t)&As[r * ldss + c];   // flat->LDS low 32 bits
        const float* gp = A + (size_t)(m0 + r) * K4 + c;
        asm volatile("global_load_async_to_lds_b128 %0, %1, off"
                     :: "v"(lds_addr), "v"(gp) : "memory");
    }
    asm volatile("s_wait_asynccnt 0" ::: "memory");
    __syncthreads();

    const int lane  = threadIdx.x & 31;
    const int wv    = threadIdx.x >> 5;
    const int n0    = wv * 16;
    const int ml    = lane & 15;          // A: M row; B/D: N col
    const int khalf = (lane >> 4) * 2;    // lanes 0-15 -> K 0,1 ; lanes 16-31 -> K 2,3

    v8f acc = {0.f, 0.f, 0.f, 0.f, 0.f, 0.f, 0.f, 0.f};
    const float* ap = &As[ml * ldss + khalf];                 // 8B aligned
    const float* bp = &Wt[(size_t)(n0 + ml) * K4 + khalf];    // 8B aligned, contiguous stream
    for (int k = 0; k < K4; k += 4) {
        v2f a = *(const v2f*)ap;  ap += 4;
        v2f b = *(const v2f*)bp;  bp += 4;
        // D = A(16x4,f32) * B(4x16,f32) + C ; full fp32 matrix pipe
        acc = __builtin_amdgcn_wmma_f32_16x16x4_f32(false, a, false, b,
                                                    (short)0, acc, false, false);
    }
    const int chi = lane >> 4;
#pragma unroll
    for (int v = 0; v < 8; ++v)
        Y[(size_t)(m0 + v + 8 * chi) * HDIM + n0 + ml] = acc[v];
}

// hout[n,c] = xw[n,c]*dinv[n]^2 + bias[c]   (self-loop + bias seed before edge scatter)
__global__ void k_init_hout(const float* __restrict__ xw, const float* __restrict__ dinv,
                            const float* __restrict__ bias, float* __restrict__ hout) {
    long i = (long)blockIdx.x * blockDim.x + threadIdx.x;
    if (i >= (long)N_NODES * HDIM) return;
    int n = (int)(i / HDIM), c = (int)(i - (long)n * HDIM);
    float di = dinv[n];
    hout[i] = xw[i] * di * di + bias[c];
}

// hout[col] += dinv[row]*dinv[col] * xw[row,:] : 32 lanes/edge, 4 channels/lane
__global__ void k_scatter(const float* __restrict__ xw, const int* __restrict__ row,
                          const int* __restrict__ col, const float* __restrict__ dinv,
                          float* __restrict__ hout) {
    long t = (long)blockIdx.x * blockDim.x + threadIdx.x;
    long e = t >> 5;
    if (e >= N_EDGES) return;
    int lane = (int)(t & 31);
    int r = row[e], cd = col[e];
    float nrm = dinv[r] * dinv[cd];
    const float4 v = *(const float4*)(xw + (size_t)r * HDIM + lane * 4);
    float* dst = hout + (size_t)cd * HDIM + lane * 4;
    atomicAdd(dst + 0, nrm * v.x);
    atomicAdd(dst + 1, nrm * v.y);
    atomicAdd(dst + 2, nrm * v.z);
    atomicAdd(dst + 3, nrm * v.w);
}

// leaky relu, write into JK concat slice (stride 768)
__global__ void k_finalize(const float* __restrict__ hout, float* __restrict__ zout, int ldz) {
    long i = (long)blockIdx.x * blockDim.x + threadIdx.x;
    if (i >= (long)N_NODES * HDIM) return;
    int n = (int)(i / HDIM), c = (int)(i - (long)n * HDIM);
    float v = hout[i];
    v = (v > 0.f) ? v : 0.01f * v;
    zout[(size_t)n * ldz + c] = v;
}

// ---------------- pooling + FC head ----------------
__global__ void k_pool(const float* __restrict__ z, const int* __restrict__ start,
                       const float* __restrict__ ogt, float* __restrict__ zb) {
    int g = blockIdx.x;
    int c = blockIdx.y * blockDim.x + threadIdx.x;
    if (c > 768) return;
    if (c == 768) { zb[g * 769 + 768] = ogt[g]; return; }
    int s = start[g], e = start[g + 1];
    float cnt = fmaxf((float)(e - s), 1.0f);
    float sum = 0.f;
    for (int n = s; n < e; ++n) sum += z[(size_t)n * 768 + c];
    zb[g * 769 + c] = sum / cnt;
}

__global__ void k_bn(const float* __restrict__ zin, int C,
                     const float* __restrict__ bg, const float* __restrict__ bb,
                     float* __restrict__ zn) {
    int c = blockIdx.x * blockDim.x + threadIdx.x;
    if (c >= C) return;
    float sum = 0.f, ssq = 0.f;
    for (int r = 0; r < N_GRAPH; ++r) { float v = zin[r * C + c]; sum += v; ssq += v * v; }
    float mu = sum * (1.0f / N_GRAPH);
    float var = ssq * (1.0f / N_GRAPH) - mu * mu;   // biased, matches BatchNorm1d training
    float is = rsqrtf(var + EPSN) * bg[c];
    for (int r = 0; r < N_GRAPH; ++r) zn[r * C + c] = (zin[r * C + c] - mu) * is + bb[c];
}

__global__ void k_fc(const float* __restrict__ zn, int Cin, int Cout,
                     const float* __restrict__ W, const float* __restrict__ b,
                     float* __restrict__ out, int relu) {
    int i = blockIdx.x * blockDim.x + threadIdx.x;
    if (i >= N_GRAPH * Cout) return;
    int r = i / Cout, co = i - r * Cout;
    float s = b[co];
    for (int k = 0; k < Cin; ++k) s += zn[r * Cin + k] * W[k * Cout + co];
    if (relu) s = (s > 0.f) ? s : 0.01f * s;
    out[i] = s;
}

// ---------------- host orchestration ----------------
extern "C" void kernel_launch(void* const* d_in, const int* in_sizes, int n_in,
                              void* d_out, int out_size, void* d_ws, size_t ws_size,
                              hipStream_t stream) {
    (void)in_sizes; (void)n_in; (void)out_size; (void)ws_size;
    const float* x     = (const float*)d_in[0];
    const float* plddt = (const float*)d_in[1];
    const float* ogt   = (const float*)d_in[2];
    const int*   ei_c  = (const int*)d_in[3];
    const int*   ei_b  = (const int*)d_in[4];
    const int*   batch = (const int*)d_in[5];

    // params flatten order: dict keys sorted ('conv'<'fc'); per conv layer: W,b,gn_a,gn_b,gn_g
    const float *cW[2][3], *cb[2][3], *ca[2][3], *cbb[2][3], *cg[2][3];
    int idx = 6;
    for (int d = 0; d < 2; ++d)
        for (int l = 0; l < 3; ++l) {
            cW[d][l]  = (const float*)d_in[idx++];
            cb[d][l]  = (const float*)d_in[idx++];
            ca[d][l]  = (const float*)d_in[idx++];
            cbb[d][l] = (const float*)d_in[idx++];
            cg[d][l]  = (const float*)d_in[idx++];
        }
    // per fc layer: W,b,bn_b,bn_g
    const float *fW[3], *fb[3], *fbb[3], *fbg[3];
    for (int l = 0; l < 3; ++l) {
        fW[l]  = (const float*)d_in[idx++];
        fb[l]  = (const float*)d_in[idx++];
        fbb[l] = (const float*)d_in[idx++];
        fbg[l] = (const float*)d_in[idx++];
    }

    // workspace carve (256B aligned slices)
    char* wptr = (char*)d_ws;
    auto carve = [&](size_t nbytes) { void* p = wptr; wptr += (nbytes + 255) & ~(size_t)255; return p; };
    float* x0p    = (float*)carve((size_t)N_NODES * 516 * 4);
    float* hn     = (float*)carve((size_t)N_NODES * 516 * 4);
    float* xw     = (float*)carve((size_t)N_NODES * HDIM * 4);
    float* hout   = (float*)carve((size_t)N_NODES * HDIM * 4);
    float* zf     = (float*)carve((size_t)N_NODES * 768 * 4);
    float* dinv   = (float*)carve((size_t)N_NODES * 4);
    float* amean  = (float*)carve((size_t)N_GRAPH * 516 * 4);
    float* invstd = (float*)carve((size_t)N_GRAPH * 516 * 4);
    float* Wt     = (float*)carve((size_t)516 * HDIM * 4);
    int*   start  = (int*)  carve((size_t)(N_GRAPH + 1) * 4);
    float* zb     = (float*)carve((size_t)N_GRAPH * 769 * 4);
    float* zn     = (float*)carve((size_t)N_GRAPH * 769 * 4);
    float* t1     = (float*)carve((size_t)N_GRAPH * 512 * 4);
    float* t2     = (float*)carve((size_t)N_GRAPH * 128 * 4);

    const int TB = 256;
    k_build_x0<<<(N_NODES * 516 + TB - 1) / TB, TB, 0, stream>>>(x, plddt, x0p);
    k_starts<<<(N_NODES + TB - 1) / TB, TB, 0, stream>>>(batch, start);

    const int* eis[2] = { ei_c, ei_b };
    for (int d = 0; d < 2; ++d) {
        const int* row = eis[d];
        const int* col = eis[d] + N_EDGES;
        k_fill<<<(N_NODES + TB - 1) / TB, TB, 0, stream>>>(dinv, 1.0f, N_NODES);
        k_degree<<<(N_EDGES + TB - 1) / TB, TB, 0, stream>>>(col, dinv);
        k_rsqrt_inplace<<<(N_NODES + TB - 1) / TB, TB, 0, stream>>>(dinv, N_NODES);

        for (int l = 0; l < 3; ++l) {
            const int C    = (l == 0) ? 513 : 128;
            const int K4   = (l == 0) ? 516 : 128;     // K padded to %4==0
            const int ldh  = (l == 0) ? 516 : 768;     // source row stride
            const int ldss = (l == 0) ? 516 : 132;     // LDS stride (ldss%64==4 -> conflict-free)
            const float* Hsrc = (l == 0) ? x0p : (zf + d * 384 + (l - 1) * HDIM);

            dim3 gs(N_GRAPH, (C + TB - 1) / TB);
            k_graph_stats<<<gs, TB, 0, stream>>>(Hsrc, ldh, C, start, ca[d][l], amean, invstd);

            long tot = (long)N_NODES * K4;
            k_gn_apply<<<(unsigned)((tot + TB - 1) / TB), TB, 0, stream>>>(
                Hsrc, ldh, C, K4, batch, amean, invstd, cg[d][l], cbb[d][l], hn);

            k_padWt<<<(K4 * HDIM + TB - 1) / TB, TB, 0, stream>>>(cW[d][l], C, K4, Wt);

            k_gemm_f32_wmma<<<N_NODES / 16, 256, (size_t)16 * ldss * 4, stream>>>(
                hn, Wt, xw, K4, ldss);

            k_init_hout<<<(N_NODES * HDIM + TB - 1) / TB, TB, 0, stream>>>(
                xw, dinv, cb[d][l], hout);
            k_scatter<<<(unsigned)(((long)N_EDGES * 32 + TB - 1) / TB), TB, 0, stream>>>(
                xw, row, col, dinv, hout);
            k_finalize<<<(N_NODES * HDIM + TB - 1) / TB, TB, 0, stream>>>(
                hout, zf + d * 384 + l * HDIM, 768);
        }
    }

    dim3 gp(N_GRAPH, 4);
    k_pool<<<gp, 256, 0, stream>>>(zf, start, ogt, zb);

    int cins[3]  = { 769, 512, 128 };
    int couts[3] = { 512, 128, 1 };
    const float* fin[3] = { zb, t1, t2 };
    float* fouts[3] = { t1, t2, (float*)d_out };
    for (int l = 0; l < 3; ++l) {
        k_bn<<<(cins[l] + TB - 1) / TB, TB, 0, stream>>>(fin[l], cins[l], fbg[l], fbb[l], zn);
        k_fc<<<(N_GRAPH * couts[l] + TB - 1) / TB, TB, 0, stream>>>(
            zn, cins[l], couts[l], fW[l], fb[l], fouts[l], (l < 2) ? 1 : 0);
    }
}